// CrfDecoderABC_30193620091161
// MI455X (gfx1250) — compile-verified
//
#include <hip/hip_runtime.h>

// Problem constants (match reference)
constexpr int BB = 256;   // batch
constexpr int TT = 512;   // time
constexpr int NN = 128;   // states
constexpr int PN = 132;   // padded LDS row stride (bank-conflict-free, 8B aligned)

typedef __attribute__((ext_vector_type(2))) float v2f;
typedef __attribute__((ext_vector_type(8))) float v8f;

// ---------------------------------------------------------------------------
// Forward (log-partition) kernel.
// Grid: 16 blocks (one per group of 16 batches), 256 threads = 8 wave32.
// Wave w computes output-state columns [16w, 16w+16) of the 16x128 alpha tile.
// Per timestep: 16x128 @ 128x128 exp-space GEMM via 32x v_wmma_f32_16x16x4_f32.
// ---------------------------------------------------------------------------
__global__ __launch_bounds__(256) void crf_forward(
    const float* __restrict__ lp,      // [B,T,N] log potentials
    const float* __restrict__ trans,   // [N,N]
    const float* __restrict__ startT,  // [N]
    const float* __restrict__ endT,    // [N]
    const int*   __restrict__ lengths, // [B]
    float*       __restrict__ logZ)    // [B] out
{
    __shared__ float pbuf[2][16][PN];  // ping-pong exp(alpha - m), padded rows
    __shared__ float wred[8][16];      // per-wave partial row maxes
    __shared__ float wsum[8][16];      // per-wave partial row sums (final lse)

    const int tid  = threadIdx.x;
    const int lane = tid & 31;
    const int w    = tid >> 5;      // wave id 0..7
    const int hi   = lane >> 4;     // lane half
    const int lo   = lane & 15;
    const int jcol = w * 16 + lo;   // this lane's output-state column
    const int b0   = blockIdx.x * 16;

    // --- Preload B fragments: E[k][j] = exp(trans[k][j]), constant over t ---
    // B 4x16 f32 layout: VGPR0 = K={0|2} (lane halves), VGPR1 = K={1|3}.
    v2f bf[32];
#pragma unroll
    for (int kk = 0; kk < 32; ++kk) {
        const int k0 = kk * 4 + hi * 2;
        bf[kk].x = __expf(trans[k0 * NN + jcol]);
        bf[kk].y = __expf(trans[(k0 + 1) * NN + jcol]);
    }

    // --- Per-register row state (C/D layout: VGPR r -> M = r + 8*hi) ---
    // Emission base for row r: batch b0 + r + 8*hi, column jcol.
    const long long base0 = (long long)(b0 + hi * 8) * TT * NN + jcol;
    const float* __restrict__ lpr = lp + base0;   // row r adds r*TT*NN (imm)

    float alpha[8];
    int   lenr[8];
#pragma unroll
    for (int r = 0; r < 8; ++r) {
        lenr[r]  = lengths[b0 + r + hi * 8];
        alpha[r] = startT[jcol] + lpr[(long long)r * TT * NN];   // t = 0
    }

    float mprev[8], mf[8];

    // Row-max across all 128 states of each of the 16 batch rows:
    //  1) butterfly max within each 16-lane half (this wave's 16 columns)
    //  2) cross-wave max via LDS, broadcast back through shfl.
    auto rowreduce_max = [&](const float v[8], float out[8]) {
#pragma unroll
        for (int r = 0; r < 8; ++r) {
            float x = v[r];
            x = fmaxf(x, __shfl_xor(x, 1, 32));
            x = fmaxf(x, __shfl_xor(x, 2, 32));
            x = fmaxf(x, __shfl_xor(x, 4, 32));
            x = fmaxf(x, __shfl_xor(x, 8, 32));
            if (lo == 0) wred[w][r + hi * 8] = x;
        }
        __syncthreads();
        float mfl = wred[0][lo];                 // each lane owns row 'lo'
#pragma unroll
        for (int ww = 1; ww < 8; ++ww) mfl = fmaxf(mfl, wred[ww][lo]);
#pragma unroll
        for (int r = 0; r < 8; ++r)              // reg r needs row r + 8*hi
            out[r] = __shfl(mfl, r + hi * 8, 32);
    };

    // Store p = exp(alpha - m) for next step (bank-conflict-free via PN pad).
    auto write_p = [&](int buf) {
#pragma unroll
        for (int r = 0; r < 8; ++r) {
            pbuf[buf][r + hi * 8][jcol] = __expf(alpha[r] - mf[r]);
            mprev[r] = mf[r];
        }
        __syncthreads();
    };

    rowreduce_max(alpha, mf);
    write_p(0);
    int buf = 0;

    // ------------------------------- scan over t -------------------------------
    for (int t = 1; t < TT; ++t) {
        const float* __restrict__ prow = &pbuf[buf][lo][0];  // A: M = lo
        v8f c = {0.f, 0.f, 0.f, 0.f, 0.f, 0.f, 0.f, 0.f};
#pragma unroll
        for (int kk = 0; kk < 32; ++kk) {
            // A 16x4 f32 layout: float2 = K {4kk+2hi, 4kk+2hi+1}
            v2f a = *(const v2f*)(prow + kk * 4 + hi * 2);
            c = __builtin_amdgcn_wmma_f32_16x16x4_f32(
                    false, a, false, bf[kk], (short)0, c, false, false);
        }

        const long long toff = (long long)t * NN;
#pragma unroll
        for (int r = 0; r < 8; ++r) {
            const float e  = lpr[toff + (long long)r * TT * NN];
            const float nv = __logf(c[r]) + mprev[r] + e;
            alpha[r] = (t < lenr[r]) ? nv : alpha[r];
            if (t + 1 < TT)  // prefetch next timestep's emissions
                __builtin_prefetch(&lpr[toff + NN + (long long)r * TT * NN], 0, 0);
        }

        rowreduce_max(alpha, mf);
        write_p(buf ^ 1);
        buf ^= 1;
    }

    // --------------------- final logsumexp(alpha + end) ---------------------
    float v[8];
#pragma unroll
    for (int r = 0; r < 8; ++r) v[r] = alpha[r] + endT[jcol];
    rowreduce_max(v, mf);
#pragma unroll
    for (int r = 0; r < 8; ++r) {
        float s = __expf(v[r] - mf[r]);
        s += __shfl_xor(s, 1, 32);
        s += __shfl_xor(s, 2, 32);
        s += __shfl_xor(s, 4, 32);
        s += __shfl_xor(s, 8, 32);
        if (lo == 0) wsum[w][r + hi * 8] = s;
    }
    __syncthreads();
    if (tid < 16) {
        float tot = 0.f, mfr = wred[0][tid];
#pragma unroll
        for (int ww = 0; ww < 8; ++ww) tot += wsum[ww][tid];
#pragma unroll
        for (int ww = 1; ww < 8; ++ww) mfr = fmaxf(mfr, wred[ww][tid]);
        logZ[b0 + tid] = mfr + __logf(tot);
    }
}

// ---------------------------------------------------------------------------
// Gold-path score + output: out[b] = score[b] - logZ[b].
// One block per batch; threads stride over T.
// ---------------------------------------------------------------------------
__global__ __launch_bounds__(256) void crf_score(
    const float* __restrict__ lp,
    const float* __restrict__ trans,
    const float* __restrict__ startT,
    const float* __restrict__ endT,
    const int*   __restrict__ target,
    const int*   __restrict__ lengths,
    const float* __restrict__ logZ,
    float*       __restrict__ out)
{
    __shared__ float red[256];
    const int b   = blockIdx.x;
    const int tid = threadIdx.x;
    const int len = lengths[b];
    const int*   tg  = target + b * TT;
    const float* lpb = lp + (long long)b * TT * NN;

    float s = 0.f;
    for (int t = tid; t < TT; t += 256) {
        if (t < len) {
            const int cur = tg[t];
            s += lpb[(long long)t * NN + cur];
            if (t >= 1) s += trans[tg[t - 1] * NN + cur];
        }
    }
    red[tid] = s;
    __syncthreads();
    for (int st = 128; st > 0; st >>= 1) {
        if (tid < st) red[tid] += red[tid + st];
        __syncthreads();
    }
    if (tid == 0) {
        const float sc = red[0] + startT[tg[0]] + endT[tg[len - 1]];
        out[b] = sc - logZ[b];
    }
}

// ---------------------------------------------------------------------------
extern "C" void kernel_launch(void* const* d_in, const int* in_sizes, int n_in,
                              void* d_out, int out_size, void* d_ws, size_t ws_size,
                              hipStream_t stream) {
    const float* lp      = (const float*)d_in[0];  // log_potentials [B,T,N]
    const float* trans   = (const float*)d_in[1];  // transition [N,N]
    const float* startT  = (const float*)d_in[2];  // start_transition [N]
    const float* endT    = (const float*)d_in[3];  // end_transition [N]
    const int*   target  = (const int*)d_in[4];    // target [B,T]
    const int*   lengths = (const int*)d_in[5];    // lengths [B]
    float* out  = (float*)d_out;                   // [B]
    float* logZ = (float*)d_ws;                    // [B] scratch

    crf_forward<<<BB / 16, 256, 0, stream>>>(lp, trans, startT, endT, lengths, logZ);
    crf_score<<<BB, 256, 0, stream>>>(lp, trans, startT, endT, target, lengths, logZ, out);
}